// WarpBicubic_52888227282960
// MI455X (gfx1250) — compile-verified
//
#include <hip/hip_runtime.h>
#include <hip/hip_bf16.h>

// Problem constants from the reference
#define BB 4
#define CC 3
#define HH 1080
#define WW 1920
#define BLOCK_X 128   // 1920 = 15 * 128 -> exact cover, 4 wave32s per block

// Catmull-Rom 5-tap parameters (matches _catmull_rom_params exactly)
__device__ __forceinline__ void catmull_params(float uv, float size, float inv_size,
                                               float& s0, float& s1, float& s2,
                                               float& ww0, float& ww1, float& ww2)
{
    float xy = uv * size;
    float tc = floorf(xy - 0.5f) + 0.5f;
    float f  = xy - tc;
    float f2 = f * f;
    float f3 = f2 * f;
    float w0 = f2 - 0.5f * (f3 + f);
    float w1 = 1.5f * f3 - 2.5f * f2 + 1.0f;
    float w3 = 0.5f * (f3 - f2);
    float w2 = 1.0f - w0 - w1 - w3;
    ww0 = w0;
    ww1 = w1 + w2;          // = 1 + 0.5*f*(1-f) >= 1, division below is safe
    ww2 = w3;
    s0 = (tc - 1.0f) * inv_size;
    s1 = (tc + w2 / ww1) * inv_size;
    s2 = (tc + 2.0f) * inv_size;
}

// One bilinear tap (padding_mode='zeros', align_corners=False), accumulated
// into 3 channel accumulators with the tap weight `sw` folded in.
// c0/c1/c2 are block-uniform (SGPR) bases; indices are 32-bit lane offsets so
// loads lower to global_load_b32 v, v_off, s[base].
__device__ __forceinline__ void bilinear_acc3(const float* __restrict__ c0,
                                              const float* __restrict__ c1,
                                              const float* __restrict__ c2,
                                              float ux, float uy, float sw,
                                              float& a0, float& a1, float& a2)
{
    float gx = ux * (float)WW - 0.5f;
    float gy = uy * (float)HH - 0.5f;
    float x0f = floorf(gx);
    float y0f = floorf(gy);
    float wx = gx - x0f;
    float wy = gy - y0f;
    int x0 = (int)x0f;
    int y0 = (int)y0f;
    int x1 = x0 + 1;
    int y1 = y0 + 1;

    bool vx0 = (x0 >= 0) && (x0 < WW);
    bool vx1 = (x1 >= 0) && (x1 < WW);
    bool vy0 = (y0 >= 0) && (y0 < HH);
    bool vy1 = (y1 >= 0) && (y1 < HH);

    unsigned xc0 = (unsigned)min(max(x0, 0), WW - 1);
    unsigned xc1 = (unsigned)min(max(x1, 0), WW - 1);
    unsigned yc0 = (unsigned)min(max(y0, 0), HH - 1);
    unsigned yc1 = (unsigned)min(max(y1, 0), HH - 1);

    // Zero-padding folded into weights: no divergent loads.
    float w00 = (vx0 && vy0) ? ((1.0f - wx) * (1.0f - wy)) * sw : 0.0f;
    float w10 = (vx1 && vy0) ? (wx * (1.0f - wy)) * sw : 0.0f;
    float w01 = (vx0 && vy1) ? ((1.0f - wx) * wy) * sw : 0.0f;
    float w11 = (vx1 && vy1) ? (wx * wy) * sw : 0.0f;

    unsigned r0 = yc0 * (unsigned)WW;
    unsigned r1 = yc1 * (unsigned)WW;
    unsigned i00 = r0 + xc0, i10 = r0 + xc1;
    unsigned i01 = r1 + xc0, i11 = r1 + xc1;

    a0 = fmaf(c0[i00], w00, a0);
    a0 = fmaf(c0[i10], w10, a0);
    a0 = fmaf(c0[i01], w01, a0);
    a0 = fmaf(c0[i11], w11, a0);

    a1 = fmaf(c1[i00], w00, a1);
    a1 = fmaf(c1[i10], w10, a1);
    a1 = fmaf(c1[i01], w01, a1);
    a1 = fmaf(c1[i11], w11, a1);

    a2 = fmaf(c2[i00], w00, a2);
    a2 = fmaf(c2[i10], w10, a2);
    a2 = fmaf(c2[i01], w01, a2);
    a2 = fmaf(c2[i11], w11, a2);
}

__global__ __launch_bounds__(BLOCK_X) void WarpBicubic_kernel(
    const float* __restrict__ img,   // [B,C,H,W]
    const float* __restrict__ vel,   // [B,2,H,W]
    float* __restrict__ out)         // [B,C,H,W]
{
    const int HW = HH * WW;
    // 3D grid: b and oy are block-uniform -> all plane bases live in SGPRs.
    const int ox  = blockIdx.x * BLOCK_X + threadIdx.x;   // [0,1920)
    const int oy  = blockIdx.y;                           // [0,1080)
    const int b   = blockIdx.z;                           // [0,4)
    const int pix = oy * WW + ox;                         // scalar + tid

    const float* imgb = img + (size_t)b * CC * HW;        // SALU-only
    const float* c0 = imgb;
    const float* c1 = imgb + HW;
    const float* c2 = imgb + 2 * HW;

    // gfx1250 prefetch (global_prefetch_b8): source region ~= dest pixel since
    // velocities are O(1) pixels. Warm the three channel planes while the
    // weight math below is in flight.
    __builtin_prefetch(c0 + pix, 0, 3);
    __builtin_prefetch(c1 + pix, 0, 3);
    __builtin_prefetch(c2 + pix, 0, 3);

    const float* velb = vel + (size_t)b * 2 * HW;
    float vx = velb[pix];        // coalesced b32 stream
    float vy = velb[HW + pix];   // coalesced b32 stream

    const float invW = 1.0f / (float)WW;
    const float invH = 1.0f / (float)HH;

    // grid = pixel-center uv minus velocity/size (reference semantics)
    float u = ((float)ox + 0.5f) * invW - vx * invW;
    float v = ((float)oy + 0.5f) * invH - vy * invH;

    float s0x, s1x, s2x, w0x, w1x, w2x;
    float s0y, s1y, s2y, w0y, w1y, w2y;
    catmull_params(u, (float)WW, invW, s0x, s1x, s2x, w0x, w1x, w2x);
    catmull_params(v, (float)HH, invH, s0y, s1y, s2y, w0y, w1y, w2y);

    float sw0 = w1x * w0y;
    float sw1 = w0x * w1y;
    float sw2 = w1x * w1y;
    float sw3 = w2x * w1y;
    float sw4 = w1x * w2y;
    float fm = 1.0f / (sw0 + sw1 + sw2 + sw3 + sw4);

    float a0 = 0.0f, a1 = 0.0f, a2 = 0.0f;
    bilinear_acc3(c0, c1, c2, s1x, s0y, sw0, a0, a1, a2);
    bilinear_acc3(c0, c1, c2, s0x, s1y, sw1, a0, a1, a2);
    bilinear_acc3(c0, c1, c2, s1x, s1y, sw2, a0, a1, a2);
    bilinear_acc3(c0, c1, c2, s2x, s1y, sw3, a0, a1, a2);
    bilinear_acc3(c0, c1, c2, s1x, s2y, sw4, a0, a1, a2);

    float* outb = out + (size_t)b * CC * HW + pix;
    outb[0]      = fmaxf(a0 * fm, 0.0f);   // coalesced b32 streams
    outb[HW]     = fmaxf(a1 * fm, 0.0f);
    outb[2 * HW] = fmaxf(a2 * fm, 0.0f);
}

extern "C" void kernel_launch(void* const* d_in, const int* in_sizes, int n_in,
                              void* d_out, int out_size, void* d_ws, size_t ws_size,
                              hipStream_t stream)
{
    const float* img = (const float*)d_in[0];   // image_prev [B,C,H,W] f32
    const float* vel = (const float*)d_in[1];   // vel_image  [B,2,H,W] f32
    float* out = (float*)d_out;                 // [B,C,H,W] f32

    dim3 block(BLOCK_X, 1, 1);                  // 4 wave32s
    dim3 grid(WW / BLOCK_X, HH, BB);            // exact cover: 15 x 1080 x 4
    WarpBicubic_kernel<<<grid, block, 0, stream>>>(img, vel, out);
}